// WSDAN_59579786330416
// MI455X (gfx1250) — compile-verified
//
#include <hip/hip_runtime.h>
#include <hip/hip_bf16.h>
#include <math.h>

// ---------------------------------------------------------------------------
// WS-DAN bilinear attention pooling head for MI455X (gfx1250, wave32).
// fp32 WMMA (V_WMMA_F32_16X16X4_F32) for all three GEMMs; async global->LDS
// (GLOBAL_LOAD_ASYNC_TO_LDS_B32, ASYNCcnt-fenced) for the transposing LDS
// stage-ins so B/A fragments become single ds_load_b64 with no fix-up moves.
// Problem is HBM-bound: ~155 MB / 23.3 TB/s ~= 7 us floor; 2.4 GFLOP fp32.
// ---------------------------------------------------------------------------

typedef __attribute__((ext_vector_type(2))) float v2f;  // A/B fragment: 2 VGPRs
typedef __attribute__((ext_vector_type(8))) float v8f;  // C/D fragment: 8 VGPRs

static __device__ inline v8f wmma_f32(v2f a, v2f b, v8f c) {
  return __builtin_amdgcn_wmma_f32_16x16x4_f32(false, a, false, b,
                                               (short)0, c, false, false);
}

// Per-lane async global->LDS copy (4 bytes). Tracked by ASYNCcnt.
static __device__ inline void async_g2l_b32(uint32_t lds_byte_addr,
                                            const float* gptr) {
  const uint64_t ga = (uint64_t)(uintptr_t)gptr;
  asm volatile("global_load_async_to_lds_b32 %0, %1, off"
               :: "v"(lds_byte_addr), "v"(ga) : "memory");
}
static __device__ inline void async_wait0() {
  asm volatile("s_wait_asynccnt 0x0" ::: "memory");
}
// Generic (flat) pointer to LDS: low 32 bits are the LDS byte offset.
static __device__ inline uint32_t lds_off(const void* p) {
  return (uint32_t)(uintptr_t)p;
}

// Problem constants
#define BATCH      32
#define HW         676      // 26*26
#define CCH        768
#define MATT       32
#define KCLS       24576    // MATT*CCH
#define NCLS       200

// ---------------------------------------------------------------------------
// Kernel 0: zero the per-batch sum-of-squares accumulators (ws is poisoned).
// ---------------------------------------------------------------------------
__global__ void k_zero_sq(float* __restrict__ sq) {
  if (threadIdx.x < BATCH) sq[threadIdx.x] = 0.0f;
}

// ---------------------------------------------------------------------------
// Kernel 1: attention_maps = feat7(21632x768) @ w_attn(768x32)
// 8 waves/block, one 16-row x 32-col tile per wave (two 16x16 accumulators).
// w_attn staged TRANSPOSED in LDS (wT[m][k], stride 132: b64 reads hit all
// 64 banks once) via async global->LDS scatter, in 128-K chunks.
// ---------------------------------------------------------------------------
#define K1_STRIDE 132
__global__ __launch_bounds__(256) void k_attn_gemm(
    const float* __restrict__ feat7,    // (B*HW, C) row-major
    const float* __restrict__ w_attn,   // (C, M) row-major
    float* __restrict__ attn_out) {     // (B*HW, M) row-major
  __shared__ float lds_wT[MATT * K1_STRIDE];     // 16.9 KB, wT[m][k] chunk
  const uint32_t lbase = lds_off(lds_wT);
  const int lane = threadIdx.x & 31;
  const int wave = threadIdx.x >> 5;
  const int tile = blockIdx.x * 8 + wave;        // 1352 row tiles
  const int row0 = tile * 16;
  const int row  = row0 + (lane & 15);
  const int koff = (lane >> 4) << 1;             // lanes 16..31 carry K+2
  const int n    = lane & 15;

  v8f acc0 = {};   // cols 0..15
  v8f acc1 = {};   // cols 16..31

  for (int chunk = 0; chunk < CCH / 128; ++chunk) {
    // Transposing stage-in: lds_wT[m][k] = w_attn[chunk*128+k][m]
    const float* src = w_attn + chunk * 128 * MATT;
    for (int i = threadIdx.x; i < 128 * MATT; i += 256) {
      const int k = i >> 5, m = i & 31;
      async_g2l_b32(lbase + (uint32_t)((m * K1_STRIDE + k) * 4), src + i);
    }
    async_wait0();
    __syncthreads();

    const float* arow = feat7 + (size_t)row * CCH + chunk * 128;
#pragma unroll 4
    for (int kk = 0; kk < 32; ++kk) {
      const int kA = kk * 4 + koff;              // even -> 8B aligned pairs
      v2f a  = *(const v2f*)(arow + kA);
      v2f b0 = *(const v2f*)(lds_wT + n * K1_STRIDE + kA);          // cols 0-15
      v2f b1 = *(const v2f*)(lds_wT + (16 + n) * K1_STRIDE + kA);   // cols 16-31
      acc0 = wmma_f32(a, b0, acc0);
      acc1 = wmma_f32(a, b1, acc1);
    }
    __syncthreads();
  }

  // C/D layout: VGPR r -> M=r (lanes 0-15) / M=r+8 (lanes 16-31), N=lane&15
  const int orow = row0 + ((lane >> 4) << 3);
  float* out = attn_out + (size_t)orow * MATT + (lane & 15);
#pragma unroll
  for (int r = 0; r < 8; ++r) {
    out[(size_t)r * MATT]      = acc0[r];
    out[(size_t)r * MATT + 16] = acc1[r];
  }
}

// ---------------------------------------------------------------------------
// Kernel 2: per batch, fm(32x768) = attn^T(32x676) @ feat6(676x768) / 676
// Block = one (batch, mtile): stages the transposed 16x676 A-panel once in
// LDS (async scatter, conflict-free stride 684), shared by 8 waves covering
// 8 of the 48 N-tiles. Grid = 32 batches x 2 mtiles x 6 groups = 384 blocks.
// ---------------------------------------------------------------------------
#define K2_STRIDE 684
__global__ __launch_bounds__(256) void k_bap_gemm(
    const float* __restrict__ attn,     // (B, HW, M)
    const float* __restrict__ feat6,    // (B, HW, C)
    float* __restrict__ fm) {           // (B, M, C)
  __shared__ float lds_a[16 * K2_STRIDE];        // ~43.8 KB, aT[m][k]
  const uint32_t lbase = lds_off(lds_a);
  const int lane = threadIdx.x & 31;
  const int wave = threadIdx.x >> 5;
  const int bidx = blockIdx.x / 12;
  const int rem  = blockIdx.x % 12;
  const int mt   = rem / 6;
  const int nt   = (rem % 6) * 8 + wave;
  const int koff = (lane >> 4) << 1;
  const int nc   = nt * 16 + (lane & 15);

  // Transposing stage-in: lds_a[m][k] = attn[b][k][mt*16+m]
  const float* Asrc = attn + (size_t)bidx * HW * MATT + mt * 16;
  for (int i = threadIdx.x; i < 16 * HW; i += 256) {
    const int k = i >> 4, m = i & 15;
    async_g2l_b32(lbase + (uint32_t)((m * K2_STRIDE + k) * 4),
                  Asrc + (size_t)k * MATT + m);
  }
  async_wait0();
  __syncthreads();

  const float* Bp = feat6 + (size_t)bidx * HW * CCH;
  const float* am = lds_a + (lane & 15) * K2_STRIDE;
  v8f acc = {};
  for (int kk = 0; kk < HW / 4; ++kk) {          // 169 WMMA steps
    const int k = kk * 4 + koff;
    v2f a = *(const v2f*)(am + k);               // single ds_load_b64
    v2f b;
    b.x = Bp[(size_t)k * CCH + nc];
    b.y = Bp[(size_t)(k + 1) * CCH + nc];
    acc = wmma_f32(a, b, acc);
  }

  const float inv = 1.0f / (float)HW;            // fold the /(h*w)
  const int orow = mt * 16 + ((lane >> 4) << 3);
  float* out = fm + ((size_t)bidx * MATT + orow) * CCH + nc;
#pragma unroll
  for (int r = 0; r < 8; ++r) out[(size_t)r * CCH] = acc[r] * inv;
}

// ---------------------------------------------------------------------------
// Kernel 3a: signed sqrt + per-batch sum of squares (LDS reduce + 1 atomic).
// 24576 % 256 == 0, so each block sits inside one batch.
// ---------------------------------------------------------------------------
__global__ __launch_bounds__(256) void k_ssqrt_reduce(
    const float* __restrict__ fm, float* __restrict__ s,
    float* __restrict__ sq) {
  __shared__ float red[256];
  const size_t i = (size_t)blockIdx.x * 256 + threadIdx.x;
  const int b = (int)(i / KCLS);
  const float x = fm[i];
  const float v = copysignf(sqrtf(fabsf(x) + 1e-12f), x);
  s[i] = v;
  red[threadIdx.x] = v * v;
  __syncthreads();
  for (int off = 128; off > 0; off >>= 1) {
    if (threadIdx.x < off) red[threadIdx.x] += red[threadIdx.x + off];
    __syncthreads();
  }
  if (threadIdx.x == 0) atomicAdd(&sq[b], red[0]);
}

// ---------------------------------------------------------------------------
// Kernel 3b: l2-normalize -> embeddings output (x100 folded into kernel 4).
// ---------------------------------------------------------------------------
__global__ __launch_bounds__(256) void k_normalize(
    const float* __restrict__ s, const float* __restrict__ sq,
    float* __restrict__ emb) {
  const size_t i = (size_t)blockIdx.x * 256 + threadIdx.x;
  const int b = (int)(i / KCLS);
  const float scale = rsqrtf(fmaxf(sq[b], 1e-12f));
  emb[i] = s[i] * scale;
}

// ---------------------------------------------------------------------------
// Kernel 4: logits = 100 * emb(32x24576) @ w_cls(24576x200)
// 26 tiles (2 Mtiles x 13 Ntiles, N padded 200->208 with lane guards).
// ---------------------------------------------------------------------------
__global__ __launch_bounds__(32) void k_cls_gemm(
    const float* __restrict__ emb, const float* __restrict__ w_cls,
    float* __restrict__ logits) {
  const int lane = threadIdx.x & 31;
  const int mt = blockIdx.x / 13;
  const int nt = blockIdx.x % 13;
  const int koff = (lane >> 4) << 1;
  const int arow = mt * 16 + (lane & 15);
  const int ncol = nt * 16 + (lane & 15);
  const bool nok = (ncol < NCLS);
  const float* A = emb + (size_t)arow * KCLS;

  v8f acc = {};
  for (int kk = 0; kk < KCLS / 4; ++kk) {        // 6144 WMMA steps
    const int k = kk * 4 + koff;
    v2f a = *(const v2f*)(A + k);                // aligned contiguous pair
    v2f b;
    b.x = nok ? w_cls[(size_t)k * NCLS + ncol] : 0.0f;
    b.y = nok ? w_cls[(size_t)(k + 1) * NCLS + ncol] : 0.0f;
    acc = wmma_f32(a, b, acc);
  }

  if (nok) {
    const int orow = mt * 16 + ((lane >> 4) << 3);
#pragma unroll
    for (int r = 0; r < 8; ++r)
      logits[(size_t)(orow + r) * NCLS + ncol] = acc[r] * 100.0f;
  }
}

// ---------------------------------------------------------------------------
// Launch: inputs = {feat_mixed6, feat_mixed7, w_attn, w_cls}
// d_out = [logits 32x200 | attention_maps 32x26x26x32 | embeddings 32x24576]
// ---------------------------------------------------------------------------
extern "C" void kernel_launch(void* const* d_in, const int* in_sizes, int n_in,
                              void* d_out, int out_size, void* d_ws, size_t ws_size,
                              hipStream_t stream) {
  const float* feat6  = (const float*)d_in[0];
  const float* feat7  = (const float*)d_in[1];
  const float* w_attn = (const float*)d_in[2];
  const float* w_cls  = (const float*)d_in[3];

  float* out    = (float*)d_out;
  float* logits = out;                                     // 32*200
  float* attn   = out + BATCH * NCLS;                      // 32*676*32
  float* emb    = attn + (size_t)BATCH * HW * MATT;        // 32*24576

  float* fm = (float*)d_ws;                                // 786432 f32
  float* s  = fm + (size_t)BATCH * KCLS;                   // 786432 f32
  float* sq = s + (size_t)BATCH * KCLS;                    // 32 f32

  k_zero_sq<<<1, 32, 0, stream>>>(sq);
  k_attn_gemm<<<(BATCH * HW / 16) / 8, 256, 0, stream>>>(feat7, w_attn, attn);
  k_bap_gemm<<<BATCH * 2 * 6, 256, 0, stream>>>(attn, feat6, fm);
  k_ssqrt_reduce<<<(BATCH * KCLS) / 256, 256, 0, stream>>>(fm, s, sq);
  k_normalize<<<(BATCH * KCLS) / 256, 256, 0, stream>>>(s, sq, emb);
  k_cls_gemm<<<2 * 13, 32, 0, stream>>>(emb, w_cls, logits);
}